// Hausdorff_7765300871716
// MI455X (gfx1250) — compile-verified
//
#include <hip/hip_runtime.h>
#include <hip/hip_bf16.h>

// ---------------------------------------------------------------------------
// Hausdorff edge distance on MI455X (gfx1250).
// Core O(Np*Ng) min-squared-distance runs on V_WMMA_F32_16X16X4_F32:
//   A row (16x4) = [py, px, 1, 0], B col (4x16) = [-2gy, -2gx, gy^2+gx^2, 0]
//   => D[M][N] = g^2 - 2 p.g  (exact in f32; coords <= 95)
// B operands are PREBUILT in the exact per-lane-half VGPR layout at edge
// compaction time, so the hot loop is: 1x b64 load + 1x wmma + 4x v_min3.
// Then cross-lane min, +p^2, atomicMax per pair.
// ---------------------------------------------------------------------------

typedef __attribute__((ext_vector_type(2))) float v2f;
typedef __attribute__((ext_vector_type(8))) float v8f;

#define NPAIR 32          // B*C = 8*4
#define HH 96
#define WW 96
#define HWSZ (HH * WW)    // 9216 (multiple of 16)

// ---------------- init workspace ----------------
__global__ void hd_init(float* Py, float* Px, v2f* Blo, v2f* Bhi,
                        int* cnts, unsigned int* result) {
    int idx = blockIdx.x * 256 + threadIdx.x;
    if (idx < NPAIR * HWSZ) {
        Py[idx] = 0.0f;
        Px[idx] = 0.0f;
        v2f zlo; zlo.x = 0.0f;    zlo.y = 0.0f;
        v2f zhi; zhi.x = 1.0e10f; zhi.y = 0.0f;   // padded GT column -> exactly 1e10
        Blo[idx] = zlo;
        Bhi[idx] = zhi;
    }
    if (idx < NPAIR * 2) cnts[idx] = 0;
    if (idx < NPAIR)     result[idx] = 0u;   // 0.0f bits
}

// ---------------- edge masks + compaction ----------------
__device__ __forceinline__ float nb(const float* im, int y, int x, float padv) {
    return (y < 0 || y >= HH || x < 0 || x >= WW) ? padv : im[y * WW + x];
}

__global__ void hd_edges(const float* __restrict__ pred,
                         const float* __restrict__ GT,
                         float* Py, float* Px, v2f* Blo, v2f* Bhi, int* cnts) {
    int idx = blockIdx.x * 256 + threadIdx.x;
    if (idx >= NPAIR * HWSZ) return;
    int pair = idx / HWSZ;           // pair = b*4 + c  ([B,C,H,W] flat)
    int pix  = idx - pair * HWSZ;
    int c    = pair & 3;
    int y = pix / WW, x = pix - (pix / WW) * WW;
    float padv = (c == 0) ? 1.0f : 0.0f;

    const float* ip = pred + (size_t)pair * HWSZ;
    const float* ig = GT   + (size_t)pair * HWSZ;

    float cp = ip[pix];
    float lp = 4.0f * cp - (nb(ip, y + 1, x, padv) + nb(ip, y - 1, x, padv) +
                            nb(ip, y, x + 1, padv) + nb(ip, y, x - 1, padv));
    float cg = ig[pix];
    float lg = 4.0f * cg - (nb(ig, y + 1, x, padv) + nb(ig, y - 1, x, padv) +
                            nb(ig, y, x + 1, padv) + nb(ig, y, x - 1, padv));

    float fy = (float)y, fx = (float)x;

    if (lp > 0.0f) {                  // clip(.,0,1) > 0  <=>  raw > 0
        int s = atomicAdd(&cnts[pair * 2 + 0], 1);
        Py[pair * HWSZ + s] = fy;
        Px[pair * HWSZ + s] = fx;
    }
    if (lg > 0.0f) {
        int s = atomicAdd(&cnts[pair * 2 + 1], 1);
        v2f lo; lo.x = -2.0f * fy;          lo.y = -2.0f * fx;  // K=0,1 rows
        v2f hi; hi.x = fy * fy + fx * fx;   hi.y = 0.0f;        // K=2,3 rows
        Blo[pair * HWSZ + s] = lo;
        Bhi[pair * HWSZ + s] = hi;
    }
}

// ---------------- WMMA distance core ----------------
// One wave (32 threads) handles one 16-row pred tile for one pair.
__global__ __launch_bounds__(32) void hd_wmma(
        const float* __restrict__ Py, const float* __restrict__ Px,
        const v2f* __restrict__ Blo, const v2f* __restrict__ Bhi,
        const int* __restrict__ cnts, unsigned int* result) {
    int pair = blockIdx.y;
    int np = cnts[pair * 2 + 0];
    int ng = cnts[pair * 2 + 1];
    int tileBase = blockIdx.x * 16;
    if (ng == 0 || tileBase >= np) return;   // wave-uniform exits only

    int lane = threadIdx.x;
    int half = lane >> 4;      // 0: K=0/1 holder, 1: K=2/3 holder
    int l16  = lane & 15;

    const float* py = Py + (size_t)pair * HWSZ;
    const float* px = Px + (size_t)pair * HWSZ;
    // Each lane-half reads its prebuilt B view; select base pointer once.
    const v2f* bsrc = (half ? Bhi : Blo) + (size_t)pair * HWSZ;

    // A (16x4): lanes 0-15 -> {K0=py, K1=px}; lanes 16-31 -> {K2=1, K3=0}
    int row = tileBase + l16;                 // <= 9215, in-bounds (init'd)
    float ay = py[row], ax = px[row];
    v2f A;
    A.x = half ? 1.0f : ay;
    A.y = half ? 0.0f : ax;

    v8f acc;
#pragma unroll
    for (int r = 0; r < 8; ++r) acc[r] = 1.0e30f;

    int ngTiles = (ng + 15) >> 4;
    for (int t = 0; t < ngTiles; ++t) {
        v2f Bv = bsrc[t * 16 + l16];          // 1x global_load_b64, prebuilt layout
        v8f d = {};
        d = __builtin_amdgcn_wmma_f32_16x16x4_f32(
                /*neg_a=*/false, A, /*neg_b=*/false, Bv,
                /*c_mod=*/(short)0, d, /*reuse_a=*/false, /*reuse_b=*/false);
#pragma unroll
        for (int r = 0; r < 8; ++r) acc[r] = fminf(acc[r], d[r]);
    }

    // Cross-lane min within each 16-lane half (columns of a row live there).
    // VGPR r holds row r (lanes 0-15) and row r+8 (lanes 16-31).
    float rowMin = 1.0e30f;
#pragma unroll
    for (int r = 0; r < 8; ++r) {
        float v = acc[r];
        v = fminf(v, __shfl_xor(v, 1, 32));
        v = fminf(v, __shfl_xor(v, 2, 32));
        v = fminf(v, __shfl_xor(v, 4, 32));
        v = fminf(v, __shfl_xor(v, 8, 32));
        if (l16 == r) rowMin = v;
    }

    if (l16 < 8) {
        int myRow = tileBase + l16 + 8 * half;   // M = l16 + 8*half
        if (myRow < np) {                        // mask padded pred rows
            float qy = py[myRow], qx = px[myRow];
            float val = rowMin + qy * qy + qx * qx;   // exact integer >= 0
            if (val < 0.0f) val = 0.0f;
            atomicMax(result + pair, __float_as_uint(val));  // nonneg floats: bit order == value order
        }
    }
}

// ---------------- finalize ----------------
__global__ void hd_final(const unsigned int* __restrict__ result, float* out) {
    if (threadIdx.x == 0 && blockIdx.x == 0) {
        float s = 0.0f;
        for (int p = 0; p < NPAIR; ++p) s += __uint_as_float(result[p]);
        out[0] = s / (320.0f * 320.0f);
    }
}

extern "C" void kernel_launch(void* const* d_in, const int* in_sizes, int n_in,
                              void* d_out, int out_size, void* d_ws, size_t ws_size,
                              hipStream_t stream) {
    const float* pred = (const float*)d_in[0];
    const float* GT   = (const float*)d_in[1];
    float* out = (float*)d_out;

    char* ws = (char*)d_ws;
    const size_t fBytes  = (size_t)NPAIR * HWSZ * sizeof(float);  // 1,179,648 B
    const size_t v2Bytes = (size_t)NPAIR * HWSZ * sizeof(v2f);    // 2,359,296 B
    float* Py  = (float*)(ws);
    float* Px  = (float*)(ws + fBytes);
    v2f*   Blo = (v2f*)  (ws + 2 * fBytes);
    v2f*   Bhi = (v2f*)  (ws + 2 * fBytes + v2Bytes);
    int*   cnts          = (int*)(ws + 2 * fBytes + 2 * v2Bytes);
    unsigned int* result = (unsigned int*)(ws + 2 * fBytes + 2 * v2Bytes + 256);

    int nTot = NPAIR * HWSZ;
    int blocks = (nTot + 255) / 256;

    hd_init<<<blocks, 256, 0, stream>>>(Py, Px, Blo, Bhi, cnts, result);
    hd_edges<<<blocks, 256, 0, stream>>>(pred, GT, Py, Px, Blo, Bhi, cnts);
    dim3 grid(HWSZ / 16, NPAIR);   // 576 pred tiles x 32 pairs; extra tiles exit early
    hd_wmma<<<grid, 32, 0, stream>>>(Py, Px, Blo, Bhi, cnts, result);
    hd_final<<<1, 32, 0, stream>>>(result, out);
}